// FedGSL_GCN_69320772157913
// MI455X (gfx1250) — compile-verified
//
#include <hip/hip_runtime.h>
#include <hip/hip_bf16.h>

#define N_NODES 100000
#define N_EDGES 3200000
#define IN_CH   512
#define HIDDEN  64
#define LOC_OUT 16
#define OUT_CH  40

typedef float v2f __attribute__((ext_vector_type(2)));
typedef float v8f __attribute__((ext_vector_type(8)));

static __device__ __forceinline__ void atomic_add_f32(float* p, float v) {
    __hip_atomic_fetch_add(p, v, __ATOMIC_RELAXED, __HIP_MEMORY_SCOPE_AGENT);
}

// ---- CDNA5 async global->LDS copy (ASYNCcnt-tracked), 16B per lane -------------
static __device__ __forceinline__ void async_ld_b128(unsigned int lds_off_bytes,
                                                     const void* gptr) {
    asm volatile("global_load_async_to_lds_b128 %0, %1, off"
                 :: "v"(lds_off_bytes), "v"(gptr) : "memory");
}
#if __has_builtin(__builtin_amdgcn_s_wait_asynccnt)
static __device__ __forceinline__ void wait_async0() { __builtin_amdgcn_s_wait_asynccnt(0); }
#else
static __device__ __forceinline__ void wait_async0() {
    asm volatile("s_wait_asynccnt 0x0" ::: "memory");
}
#endif
static __device__ __forceinline__ unsigned int lds_off(const void* p) {
    return (unsigned int)(uintptr_t)p;   // low 32 bits of flat shared addr = LDS offset
}

// ---------------------------------------------------------------- init / degree
__global__ __launch_bounds__(256) void init_ws(float* __restrict__ agg1,
                                               float* __restrict__ agg2,
                                               float* __restrict__ deg) {
    int i = blockIdx.x * 256 + threadIdx.x;
    if (i < N_NODES * HIDDEN)  agg1[i] = 0.0f;
    if (i < N_NODES * LOC_OUT) agg2[i] = 0.0f;
    if (i < N_NODES)           deg[i]  = 1.0f;   // self-loop
}

__global__ __launch_bounds__(256) void deg_count(const long long* __restrict__ col,
                                                 float* __restrict__ deg) {
    int e = blockIdx.x * 256 + threadIdx.x;
    if (e < N_EDGES) atomic_add_f32(&deg[(int)col[e]], 1.0f);
}

__global__ __launch_bounds__(256) void make_dinv(float* __restrict__ deg) {
    int i = blockIdx.x * 256 + threadIdx.x;
    if (i < N_NODES) deg[i] = rsqrtf(fmaxf(deg[i], 1.0f));
}

// Pre-pair weights: Wp[k/2][n][2] = {W[k][n], W[k+1][n]}  -> B-frag is one ds_load_b64
__global__ __launch_bounds__(256) void pair_w1(const float* __restrict__ W1,
                                               float* __restrict__ W1p) {
    int i = blockIdx.x * 256 + threadIdx.x;          // (512/2)*64 = 16384 pairs
    if (i >= (IN_CH / 2) * HIDDEN) return;
    int kp = i >> 6;          // /HIDDEN
    int nn = i & 63;
    W1p[kp * 128 + nn * 2 + 0] = W1[(2 * kp)     * HIDDEN + nn];
    W1p[kp * 128 + nn * 2 + 1] = W1[(2 * kp + 1) * HIDDEN + nn];
}
__global__ __launch_bounds__(256) void pair_w2(const float* __restrict__ W2,
                                               float* __restrict__ W2p) {
    int i = blockIdx.x * 256 + threadIdx.x;          // (64/2)*16 = 512 pairs
    if (i >= (HIDDEN / 2) * LOC_OUT) return;
    int kp = i >> 4;
    int nn = i & 15;
    W2p[kp * 32 + nn * 2 + 0] = W2[(2 * kp)     * LOC_OUT + nn];
    W2p[kp * 32 + nn * 2 + 1] = W2[(2 * kp + 1) * LOC_OUT + nn];
}

// ---------------------------------------------------------------- GEMM1: H = X @ W1
// fp32 WMMA 16x16x4. Block = 256 thr (8 waves), tile 128 rows x 64 cols.
// K-tiles of 32 double-buffered in LDS via async global->LDS copies.
__global__ __launch_bounds__(256) void gemm1_wmma(const float* __restrict__ X,
                                                  const float* __restrict__ W1p,
                                                  float* __restrict__ H) {
    __shared__ float As[2][128 * 36];    // 128 rows x 32 K, row stride 36 (16B aligned)
    __shared__ float Bs[2][16 * 128];    // 16 kp x (64 n x 2) paired, contiguous 8KB

    const int tid  = threadIdx.x;
    const int wave = tid >> 5;
    const int lane = tid & 31;
    const int rowBase = blockIdx.x * 128;
    const int n  = lane & 15;            // A-frag row / B,C-frag col
    const int kq = (lane >> 4) << 1;     // K sub-offset 0/2 (ISA 16x4 layout)

    // per-thread A-tile source rows/cols and LDS offsets
    int arow[4], acol[4];
    unsigned int aoff[2][4], boff[2][2];
#pragma unroll
    for (int j = 0; j < 4; ++j) {
        int i  = tid + j * 256;          // 0..1023 float4 chunks
        int r  = i >> 3;
        int c4 = i & 7;
        int gr = rowBase + r; if (gr >= N_NODES) gr = N_NODES - 1;
        arow[j] = gr; acol[j] = c4 * 4;
        aoff[0][j] = lds_off(&As[0][r * 36 + c4 * 4]);
        aoff[1][j] = lds_off(&As[1][r * 36 + c4 * 4]);
    }
#pragma unroll
    for (int j = 0; j < 2; ++j) {
        int i = tid + j * 256;           // 0..511 float4 chunks of the 8KB W tile
        boff[0][j] = lds_off(&Bs[0][i * 4]);
        boff[1][j] = lds_off(&Bs[1][i * 4]);
    }

    // prologue: issue async copies for K-tile 0 into buffer 0
#pragma unroll
    for (int j = 0; j < 4; ++j)
        async_ld_b128(aoff[0][j], X + (size_t)arow[j] * IN_CH + acol[j]);
#pragma unroll
    for (int j = 0; j < 2; ++j)
        async_ld_b128(boff[0][j], W1p + (tid + j * 256) * 4);

    v8f acc[4] = {v8f{0}, v8f{0}, v8f{0}, v8f{0}};

    for (int kk = 0; kk < IN_CH; kk += 32) {
        const int p = (kk >> 5) & 1;
        wait_async0();                   // my async writes into buffer p landed
        __syncthreads();                 // everyone's landed; prev compute done

        if (kk + 32 < IN_CH) {           // prefetch next K-tile into other buffer
            const int q = p ^ 1;
            const int kn = kk + 32;
#pragma unroll
            for (int j = 0; j < 4; ++j)
                async_ld_b128(aoff[q][j], X + (size_t)arow[j] * IN_CH + kn + acol[j]);
#pragma unroll
            for (int j = 0; j < 2; ++j)
                async_ld_b128(boff[q][j], W1p + (kn >> 1) * 128 + (tid + j * 256) * 4);
        }

        const float* Ap = &As[p][(wave * 16 + n) * 36];
        const float* Bp = &Bs[p][n * 2];
#pragma unroll
        for (int k0 = 0; k0 < 32; k0 += 4) {
            v2f a = *(const v2f*)&Ap[k0 + kq];
            const int kpo = ((k0 + kq) >> 1) * 128;
#pragma unroll
            for (int j = 0; j < 4; ++j) {
                v2f b = *(const v2f*)&Bp[kpo + j * 32];
                acc[j] = __builtin_amdgcn_wmma_f32_16x16x4_f32(
                    false, a, false, b, (short)0, acc[j], false, false);
            }
        }
        __syncthreads();                 // all waves done with buffer p
    }

    const int mh = (lane >> 4) * 8;      // C/D layout: VGPR v -> M = v (+8 upper lanes)
#pragma unroll
    for (int j = 0; j < 4; ++j)
#pragma unroll
        for (int v = 0; v < 8; ++v) {
            int gr = rowBase + wave * 16 + v + mh;
            if (gr < N_NODES) H[(size_t)gr * HIDDEN + j * 16 + n] = acc[j][v];
        }
}

// ---------------------------------------------------------------- edge aggregation #1
__global__ __launch_bounds__(256) void agg1_edges(const long long* __restrict__ ei,
                                                  const float* __restrict__ H,
                                                  const float* __restrict__ dinv,
                                                  float* __restrict__ agg) {
    int gid = blockIdx.x * 256 + threadIdx.x;
    int e = gid >> 5;
    if (e >= N_EDGES) return;
    int c = (gid & 31) * 2;
    int s = (int)ei[e];
    int t = (int)ei[N_EDGES + e];
    float nrm = dinv[s] * dinv[t];
    float2 hv = *(const float2*)&H[(size_t)s * HIDDEN + c];
    atomic_add_f32(&agg[(size_t)t * HIDDEN + c],     hv.x * nrm);
    atomic_add_f32(&agg[(size_t)t * HIDDEN + c + 1], hv.y * nrm);
}

// self-loop + bias + relu, in place over agg1 -> x1
__global__ __launch_bounds__(256) void fuse1(const float* __restrict__ H,
                                             const float* __restrict__ dinv,
                                             const float* __restrict__ b1,
                                             float* __restrict__ agg) {
    int i = blockIdx.x * 256 + threadIdx.x;
    if (i >= N_NODES * HIDDEN) return;
    int node = i >> 6;
    int c    = i & 63;
    float di = dinv[node];
    float v  = agg[i] + H[i] * di * di + b1[c];
    agg[i] = fmaxf(v, 0.0f);
}

// ---------------------------------------------------------------- GEMM2: H2 = x1 @ W2
__global__ __launch_bounds__(256) void gemm2_wmma(const float* __restrict__ X1,
                                                  const float* __restrict__ W2p,
                                                  float* __restrict__ H2) {
    __shared__ float Bs[(HIDDEN / 2) * LOC_OUT * 2];   // 1024 floats, paired layout
    const int tid = threadIdx.x;
    ((float4*)Bs)[tid] = ((const float4*)W2p)[tid];    // 256 float4 flat copy
    __syncthreads();

    const int wave = tid >> 5, lane = tid & 31;
    const int n  = lane & 15;
    const int kq = (lane >> 4) << 1;
    int r0 = blockIdx.x * 128 + wave * 16 + n;
    int gr = r0 < N_NODES ? r0 : N_NODES - 1;

    v8f acc = v8f{0};
#pragma unroll
    for (int k0 = 0; k0 < HIDDEN; k0 += 4) {
        v2f a = *(const v2f*)&X1[(size_t)gr * HIDDEN + k0 + kq];
        v2f b = *(const v2f*)&Bs[((k0 + kq) >> 1) * 32 + n * 2];
        acc = __builtin_amdgcn_wmma_f32_16x16x4_f32(
            false, a, false, b, (short)0, acc, false, false);
    }
    const int mh = (lane >> 4) * 8;
#pragma unroll
    for (int v = 0; v < 8; ++v) {
        int gw = blockIdx.x * 128 + wave * 16 + v + mh;
        if (gw < N_NODES) H2[(size_t)gw * LOC_OUT + n] = acc[v];
    }
}

// ---------------------------------------------------------------- edge aggregation #2
__global__ __launch_bounds__(256) void agg2_edges(const long long* __restrict__ ei,
                                                  const float* __restrict__ H2,
                                                  const float* __restrict__ dinv,
                                                  float* __restrict__ agg) {
    int gid = blockIdx.x * 256 + threadIdx.x;
    int e = gid >> 3;
    if (e >= N_EDGES) return;
    int c = (gid & 7) * 2;
    int s = (int)ei[e];
    int t = (int)ei[N_EDGES + e];
    float nrm = dinv[s] * dinv[t];
    float2 hv = *(const float2*)&H2[(size_t)s * LOC_OUT + c];
    atomic_add_f32(&agg[(size_t)t * LOC_OUT + c],     hv.x * nrm);
    atomic_add_f32(&agg[(size_t)t * LOC_OUT + c + 1], hv.y * nrm);
}

// ---------------------------------------------------------------- fused head
__global__ __launch_bounds__(256) void final_head(const float* __restrict__ H2,
                                                  const float* __restrict__ agg2,
                                                  const float* __restrict__ dinv,
                                                  const float* __restrict__ b2,
                                                  const float* __restrict__ glob,
                                                  const float* __restrict__ aw1,
                                                  const float* __restrict__ ab1,
                                                  const float* __restrict__ aw2,
                                                  const float* __restrict__ lw,
                                                  const float* __restrict__ lb,
                                                  float* __restrict__ out) {
    int nd = blockIdx.x * 256 + threadIdx.x;
    if (nd >= N_NODES) return;

    float z0[16], z1[16];
    float di = dinv[nd];
    float dd = di * di;
#pragma unroll
    for (int c = 0; c < 16; ++c) {
        z0[c] = agg2[(size_t)nd * 16 + c] + H2[(size_t)nd * 16 + c] * dd + b2[c];
        z1[c] = glob[(size_t)nd * 16 + c];
    }
    float w0 = 0.0f, w1 = 0.0f;
#pragma unroll
    for (int i = 0; i < 16; ++i) {
        float s0 = ab1[i], s1 = ab1[i];
#pragma unroll
        for (int j = 0; j < 16; ++j) {
            float wji = aw1[j * 16 + i];
            s0 += z0[j] * wji;
            s1 += z1[j] * wji;
        }
        float a2 = aw2[i];
        w0 += tanhf(s0) * a2;
        w1 += tanhf(s1) * a2;
    }
    float m  = fmaxf(w0, w1);
    float e0 = expf(w0 - m), e1 = expf(w1 - m);
    float be0 = e0 / (e0 + e1);
    float be1 = 1.0f - be0;

    float emb[16];
#pragma unroll
    for (int c = 0; c < 16; ++c) emb[c] = be0 * z0[c] + be1 * z1[c];

#pragma unroll
    for (int o = 0; o < OUT_CH; ++o) {
        float s = lb[o];
#pragma unroll
        for (int c = 0; c < 16; ++c) s += emb[c] * lw[c * OUT_CH + o];
        out[(size_t)nd * OUT_CH + o] = s;
    }
}

// ---------------------------------------------------------------- launch
extern "C" void kernel_launch(void* const* d_in, const int* in_sizes, int n_in,
                              void* d_out, int out_size, void* d_ws, size_t ws_size,
                              hipStream_t stream) {
    const float*     X   = (const float*)d_in[0];
    const long long* EI  = (const long long*)d_in[1];   // int64 [2, E]
    const float*     G   = (const float*)d_in[2];
    const float*     W1  = (const float*)d_in[3];
    const float*     B1  = (const float*)d_in[4];
    const float*     W2  = (const float*)d_in[5];
    const float*     B2  = (const float*)d_in[6];
    const float*     AW1 = (const float*)d_in[7];
    const float*     AB1 = (const float*)d_in[8];
    const float*     AW2 = (const float*)d_in[9];
    const float*     LW  = (const float*)d_in[10];
    const float*     LB  = (const float*)d_in[11];
    float* out = (float*)d_out;

    float* deg  = (float*)d_ws;                         // N
    float* h1   = deg  + N_NODES;                       // N*64
    float* agg1 = h1   + (size_t)N_NODES * HIDDEN;      // N*64 (reused as x1)
    float* h2   = agg1 + (size_t)N_NODES * HIDDEN;      // N*16
    float* agg2 = h2   + (size_t)N_NODES * LOC_OUT;     // N*16
    float* w1p  = agg2 + (size_t)N_NODES * LOC_OUT;     // 512*64
    float* w2p  = w1p  + (size_t)IN_CH * HIDDEN;        // 64*16

    (void)in_sizes; (void)n_in; (void)out_size; (void)ws_size;

    init_ws  <<<(N_NODES * HIDDEN + 255) / 256, 256, 0, stream>>>(agg1, agg2, deg);
    deg_count<<<(N_EDGES + 255) / 256,          256, 0, stream>>>(EI + N_EDGES, deg);
    make_dinv<<<(N_NODES + 255) / 256,          256, 0, stream>>>(deg);
    pair_w1  <<<((IN_CH / 2) * HIDDEN + 255) / 256, 256, 0, stream>>>(W1, w1p);
    pair_w2  <<<((HIDDEN / 2) * LOC_OUT + 255) / 256, 256, 0, stream>>>(W2, w2p);

    gemm1_wmma<<<(N_NODES + 127) / 128, 256, 0, stream>>>(X, w1p, h1);
    agg1_edges<<<(N_EDGES * 32 + 255) / 256, 256, 0, stream>>>(EI, h1, deg, agg1);
    fuse1     <<<(N_NODES * HIDDEN + 255) / 256, 256, 0, stream>>>(h1, deg, B1, agg1);

    gemm2_wmma<<<(N_NODES + 127) / 128, 256, 0, stream>>>(agg1, w2p, h2);
    agg2_edges<<<(N_EDGES * 8 + 255) / 256, 256, 0, stream>>>(EI, h2, deg, agg2);

    final_head<<<(N_NODES + 255) / 256, 256, 0, stream>>>(h2, agg2, deg, B2, G,
                                                          AW1, AB1, AW2, LW, LB, out);
}